// ConditionalTimeGradPredictionNetwork_29540785062134
// MI455X (gfx1250) — compile-verified
//
#include <hip/hip_runtime.h>
#include <math.h>

// ---------------------------------------------------------------------------
// MI455X (gfx1250) pipeline for ConditionalTimeGradPredictionNetwork.
// VALU-bound workload (normcdfinv on 33.5M elems) => maximize wave parallelism
// by segmenting the EWMA scan (linear recurrence) into S independent pieces.
//   k1: per-segment plain/weighted sums  (V_WMMA_F32_16X16X4_F32)
//   k2: sequential segment combine, loc, scale_future, forecasts
//   k3: per-segment scan + t4-CDF + ndtri (async global->LDS double buffer)
// Fallback single fused kernel if the harness scratch is too small.
// ---------------------------------------------------------------------------

#define GAS __attribute__((address_space(1)))
#define LAS __attribute__((address_space(3)))

typedef float v2f __attribute__((ext_vector_type(2)));
typedef float v8f __attribute__((ext_vector_type(8)));
typedef int   v4i __attribute__((ext_vector_type(4)));

namespace {
constexpr int   kB     = 128;
constexpr int   kT     = 2048;
constexpr int   kP     = 64;
constexpr int   kD     = 128;
constexpr int   kTOut  = kT + kP;       // 2112
constexpr int   kS     = 8;             // scan segments per series
constexpr int   kL     = kT / kS;       // 256 timesteps per segment
constexpr float kAlpha = 0.94f;
constexpr float kEps   = 1e-5f;
constexpr float kUClip = 1e-6f;
constexpr int   kCT    = 32;            // staged rows per chunk (16 KB)
constexpr int   kRPW   = kCT / 4;       // async b128 ops per wave per chunk

// workspace layout (floats)
constexpr size_t kSeg     = (size_t)kB * kS * kD;
constexpr size_t kOffSum  = 0;
constexpr size_t kOffW1   = kOffSum + kSeg;
constexpr size_t kOffW2   = kOffW1 + kSeg;
constexpr size_t kOffVIn  = kOffW2 + kSeg;
constexpr size_t kOffLoc  = kOffVIn + kSeg;
constexpr size_t kWsFloats = kOffLoc + (size_t)kB * kD;
constexpr size_t kWsBytes  = kWsFloats * sizeof(float);
}

__device__ __forceinline__ float student_t4_u(float tt) {
  // u = 0.5 + ss*(0.75 - 0.25*ss^2),  ss = tt / sqrt(4 + tt^2)
  float ss = tt * rsqrtf(fmaf(tt, tt, 4.0f));
  float u  = fmaf(ss, fmaf(-0.25f, ss * ss, 0.75f), 0.5f);
  return fminf(fmaxf(u, kUClip), 1.0f - kUClip);
}

__device__ __forceinline__ float t4_ppf_scaled(float z, float scale_f, float loc) {
  float u  = fminf(fmaxf(normcdff(z), kUClip), 1.0f - kUClip);
  float a4 = fminf(fmaxf(4.0f * u * (1.0f - u), kUClip), 1.0f);
  float sa    = sqrtf(a4);
  float inner = cosf(acosf(sa) * (1.0f / 3.0f)) / sa - 1.0f;
  float q     = 2.0f * sqrtf(fmaxf(inner, 0.0f));
  float sgn   = (u > 0.5f) ? 1.0f : ((u < 0.5f) ? -1.0f : 0.0f);
  return fmaf(sgn * q, scale_f, loc);
}

#define WMMA_F32X4(A, Bm, C) \
  __builtin_amdgcn_wmma_f32_16x16x4_f32(false, (A), false, (Bm), (short)0, (C), false, false)

// ====================== k1: per-segment sums (WMMA) ======================
// Wave w covers d in [32w, 32w+32) (2 groups of 16), full segment in t.
// A(16x4): lane 0-15 M=lane K0/K1 in vgpr0/1; lane 16-31 same M, K2/K3.
// B(4x16): value depends only on K; assumed K split mirrors A (K0/K1 low
// lanes' vgpr0/1, K2/K3 high lanes). For B = ones the layout is irrelevant.
__global__ __launch_bounds__(128)
void k1_segment_stats(const float* __restrict__ x_hist, float* __restrict__ ws)
{
  const int blk = blockIdx.x;                 // b * kS + s
  const int tid = threadIdx.x, lane = tid & 31, wave = tid >> 5;
  const float* xs = x_hist + (size_t)blk * kL * kD;  // b*T*D + s*L*D

  __shared__ float shS[kD], sh1[kD], sh2[kD];

  const int laneM = lane & 15;
  const int laneK = (lane >> 4) << 1;
  const float r  = 1.0f / kAlpha;
  const float r4 = (r * r) * (r * r);

  v8f aS0 = {}, aS1 = {}, a10 = {}, a11 = {}, a20 = {}, a21 = {};
  const v2f ones = {1.0f, 1.0f};
  v2f bw;                                     // alpha^{-j} weights for this lane
  bw.x = (lane < 16) ? 1.0f : r * r;
  bw.y = bw.x * r;

  const int d0 = 32 * wave;
  for (int j0 = 0; j0 < kL; j0 += 4) {
    const float* rp = xs + (size_t)(j0 + laneK) * kD + laneM;
    v2f a0 = {rp[d0],      rp[kD + d0]};
    v2f a1 = {rp[d0 + 16], rp[kD + d0 + 16]};
    v2f q0 = a0 * a0, q1 = a1 * a1;
    aS0 = WMMA_F32X4(a0, ones, aS0);
    aS1 = WMMA_F32X4(a1, ones, aS1);
    a10 = WMMA_F32X4(a0, bw,   a10);
    a11 = WMMA_F32X4(a1, bw,   a11);
    a20 = WMMA_F32X4(q0, bw,   a20);
    a21 = WMMA_F32X4(q1, bw,   a21);
    bw.x *= r4; bw.y *= r4;
  }
  // D(16x16): lane n vgpr r -> D[r,n]; lane n+16 vgpr r -> D[8+r,n].
  if (lane == 0) {
    #pragma unroll
    for (int rr = 0; rr < 8; ++rr) {
      shS[d0 + rr] = aS0[rr];  shS[d0 + 16 + rr] = aS1[rr];
      sh1[d0 + rr] = a10[rr];  sh1[d0 + 16 + rr] = a11[rr];
      sh2[d0 + rr] = a20[rr];  sh2[d0 + 16 + rr] = a21[rr];
    }
  } else if (lane == 16) {
    #pragma unroll
    for (int rr = 0; rr < 8; ++rr) {
      shS[d0 + 8 + rr] = aS0[rr];  shS[d0 + 24 + rr] = aS1[rr];
      sh1[d0 + 8 + rr] = a10[rr];  sh1[d0 + 24 + rr] = a11[rr];
      sh2[d0 + 8 + rr] = a20[rr];  sh2[d0 + 24 + rr] = a21[rr];
    }
  }
  __syncthreads();
  const size_t o = (size_t)blk * kD + tid;
  ws[kOffSum + o] = shS[tid];
  ws[kOffW1  + o] = sh1[tid];
  ws[kOffW2  + o] = sh2[tid];
}

// ====================== k2: combine + forecasts ======================
__global__ __launch_bounds__(256)
void k2_combine(const float* __restrict__ x_hist,
                const float* __restrict__ z_future,
                float* __restrict__ out, float* __restrict__ ws)
{
  const int idx = blockIdx.x * 256 + threadIdx.x;   // b*kD + d
  if (idx >= kB * kD) return;
  const int b = idx / kD, d = idx - b * kD;

  float tot = 0.0f;
  #pragma unroll
  for (int s = 0; s < kS; ++s)
    tot += ws[kOffSum + ((size_t)(b * kS + s)) * kD + d];
  const float loc = tot * (1.0f / (float)kT);
  ws[kOffLoc + (size_t)b * kD + d] = loc;

  const float r    = 1.0f / kAlpha;
  const float w0   = (powf(r, (float)kL) - 1.0f) / (r - 1.0f); // sum r^j
  const float aLm1 = powf(kAlpha, (float)(kL - 1));
  const float aL   = aLm1 * kAlpha;

  const float x0  = x_hist[(size_t)b * kT * kD + d];
  const float dv0 = x0 - loc;
  const float c20 = dv0 * dv0;

  // var_end[s] = aL*var_in[s] + (1-a)*a^{L-1}*(W2 - 2 loc W1 + loc^2 W0)
  // segment 0: ewma[0] = c2[0] => add (a^{L-1} - (1-a)a^{L-1}) c20 = aL*c20.
  float var = 0.0f;
  #pragma unroll
  for (int s = 0; s < kS; ++s) {
    const size_t o = ((size_t)(b * kS + s)) * kD + d;
    ws[kOffVIn + o] = var;
    const float w1 = ws[kOffW1 + o], w2 = ws[kOffW2 + o];
    const float Rg = (1.0f - kAlpha) * aLm1 *
                     (w2 - 2.0f * loc * w1 + loc * loc * w0);
    var = (s == 0) ? (Rg + aL * c20) : fmaf(aL, var, Rg);
  }

  const float scale_f = fmaxf(sqrtf(var * 0.5f), kEps);
  const float* zb  = z_future + (size_t)b * kP * kD + d;
  float*       ofc = out + (size_t)b * kTOut * kD + (size_t)kT * kD + d;
  for (int p = 0; p < kP; ++p)
    ofc[(size_t)p * kD] = t4_ppf_scaled(zb[(size_t)p * kD], scale_f, loc);
}

// ====================== k3: per-segment scan + emit ======================
__global__ __launch_bounds__(128)
void k3_emit(const float* __restrict__ x_hist, const float* __restrict__ ws,
             float* __restrict__ out)
{
  const int blk = blockIdx.x;                 // b * kS + s
  const int b = blk / kS, s = blk - b * kS;
  const int tid = threadIdx.x, lane = tid & 31, wave = tid >> 5;
  const float* xs = x_hist + (size_t)blk * kL * kD;
  float* ocol = out + (size_t)b * kTOut * kD + (size_t)s * kL * kD + tid;

  __shared__ float buf[2][kCT * kD];          // 2 x 16 KB streaming buffers

  const float loc = ws[kOffLoc + (size_t)b * kD + tid];
  float var = ws[kOffVIn + (size_t)blk * kD + tid];

  auto issue_chunk = [&](int c) {
    const int bi = c & 1;
    #pragma unroll
    for (int i = 0; i < kRPW; ++i) {
      const int rr = wave * kRPW + i;
      GAS char* g = (GAS char*)(xs + (size_t)(c * kCT + rr) * kD) + lane * 16;
      LAS char* l = (LAS char*)(&buf[bi][rr * kD]) + lane * 16;
      __builtin_amdgcn_global_load_async_to_lds_b128((GAS v4i*)g, (LAS v4i*)l,
                                                     0, 0);
    }
  };
  issue_chunk(0);
  issue_chunk(1);
  const bool seg0 = (s == 0);
  constexpr int kNCh = kL / kCT;              // 8 chunks
  for (int c = 0; c < kNCh; ++c) {
    if (c + 1 < kNCh) __builtin_amdgcn_s_wait_asynccnt(kRPW);
    else              __builtin_amdgcn_s_wait_asynccnt(0);
    __syncthreads();
    const float* bp = &buf[c & 1][tid];
    #pragma unroll 4
    for (int rr = 0; rr < kCT; ++rr) {
      const int j = c * kCT + rr;
      float x  = bp[rr * kD];
      float dv = x - loc;
      float c2 = dv * dv;
      var = (seg0 && j == 0) ? c2 : fmaf(kAlpha, var, (1.0f - kAlpha) * c2);
      float scale = fmaxf(sqrtf(var * 0.5f), kEps);  // (DF-2)/DF = 0.5
      ocol[(size_t)j * kD] = normcdfinvf(student_t4_u(dv / scale));
    }
    __syncthreads();
    if (c + 2 < kNCh) issue_chunk(c + 2);
  }
}

// ============ fallback: round-2 fused kernel (small scratch) ============
__global__ __launch_bounds__(128)
void tgrad_fused_kernel(const float* __restrict__ x_hist,
                        const float* __restrict__ z_future,
                        float* __restrict__ out)
{
  const int b    = blockIdx.x;
  const int tid  = threadIdx.x;
  const int lane = tid & 31;
  const int wave = tid >> 5;
  const float* xb = x_hist + (size_t)b * kT * kD;

  __shared__ float part[4][kD];
  __shared__ float buf[2][kCT * kD];

  float loc;
  {
    const int laneM = lane & 15;
    const int laneK = (lane >> 4) << 1;
    v8f acc[8] = {};
    const v2f ones = {1.0f, 1.0f};
    const int t0b = wave * (kT / 4);
    for (int t0 = t0b; t0 < t0b + kT / 4; t0 += 4) {
      const float* rp = xb + (size_t)(t0 + laneK) * kD + laneM;
      #pragma unroll
      for (int g = 0; g < 8; ++g) {
        v2f a = {rp[g * 16], rp[kD + g * 16]};
        acc[g] = WMMA_F32X4(a, ones, acc[g]);
      }
    }
    if (lane == 0) {
      #pragma unroll
      for (int g = 0; g < 8; ++g)
        #pragma unroll
        for (int rr = 0; rr < 8; ++rr) part[wave][g * 16 + rr] = acc[g][rr];
    } else if (lane == 16) {
      #pragma unroll
      for (int g = 0; g < 8; ++g)
        #pragma unroll
        for (int rr = 0; rr < 8; ++rr) part[wave][g * 16 + 8 + rr] = acc[g][rr];
    }
    __syncthreads();
    loc = (part[0][tid] + part[1][tid] + part[2][tid] + part[3][tid]) *
          (1.0f / (float)kT);
  }

  float var = 0.0f;
  float* ocol = out + (size_t)b * kTOut * kD + tid;
  auto issue_chunk = [&](int c) {
    const int bi = c & 1;
    #pragma unroll
    for (int i = 0; i < kRPW; ++i) {
      const int rr = wave * kRPW + i;
      GAS char* g = (GAS char*)(xb + (size_t)(c * kCT + rr) * kD) + lane * 16;
      LAS char* l = (LAS char*)(&buf[bi][rr * kD]) + lane * 16;
      __builtin_amdgcn_global_load_async_to_lds_b128((GAS v4i*)g, (LAS v4i*)l,
                                                     0, 0);
    }
  };
  issue_chunk(0);
  issue_chunk(1);
  constexpr int kNChF = kT / kCT;
  for (int c = 0; c < kNChF; ++c) {
    if (c + 1 < kNChF) __builtin_amdgcn_s_wait_asynccnt(kRPW);
    else               __builtin_amdgcn_s_wait_asynccnt(0);
    __syncthreads();
    const float* bp = &buf[c & 1][tid];
    #pragma unroll 4
    for (int rr = 0; rr < kCT; ++rr) {
      const int t = c * kCT + rr;
      float x  = bp[rr * kD];
      float dv = x - loc;
      float c2 = dv * dv;
      var = (t == 0) ? c2 : fmaf(kAlpha, var, (1.0f - kAlpha) * c2);
      float scale = fmaxf(sqrtf(var * 0.5f), kEps);
      ocol[(size_t)t * kD] = normcdfinvf(student_t4_u(dv / scale));
    }
    __syncthreads();
    if (c + 2 < kNChF) issue_chunk(c + 2);
  }

  {
    const float scale_f = fmaxf(sqrtf(var * 0.5f), kEps);
    const float* zb  = z_future + (size_t)b * kP * kD;
    float*       ofc = out + (size_t)b * kTOut * kD + (size_t)kT * kD + tid;
    for (int p = 0; p < kP; ++p)
      ofc[(size_t)p * kD] = t4_ppf_scaled(zb[(size_t)p * kD + tid], scale_f, loc);
  }
}

extern "C" void kernel_launch(void* const* d_in, const int* in_sizes, int n_in,
                              void* d_out, int out_size, void* d_ws,
                              size_t ws_size, hipStream_t stream) {
  (void)in_sizes; (void)n_in; (void)out_size;
  const float* x_hist   = (const float*)d_in[0];
  const float* z_future = (const float*)d_in[1];
  float*       out      = (float*)d_out;
  float*       ws       = (float*)d_ws;

  if (d_ws != nullptr && ws_size >= kWsBytes) {
    hipLaunchKernelGGL(k1_segment_stats, dim3(kB * kS), dim3(kD), 0, stream,
                       x_hist, ws);
    hipLaunchKernelGGL(k2_combine, dim3((kB * kD) / 256), dim3(256), 0, stream,
                       x_hist, z_future, out, ws);
    hipLaunchKernelGGL(k3_emit, dim3(kB * kS), dim3(kD), 0, stream,
                       x_hist, ws, out);
  } else {
    hipLaunchKernelGGL(tgrad_fused_kernel, dim3(kB), dim3(kD), 0, stream,
                       x_hist, z_future, out);
  }
}